// NeuralODESystem_31516470018621
// MI455X (gfx1250) — compile-verified
//
#include <hip/hip_runtime.h>
#include <hip/hip_bf16.h>

typedef __attribute__((ext_vector_type(2))) float v2f;
typedef __attribute__((ext_vector_type(8))) float v8f;

// Problem constants (match reference setup_inputs)
constexpr int B_ = 4096, D_ = 64, H_ = 256, T_ = 50, SUB = 4;
constexpr int MT  = 32;    // batch rows per workgroup
constexpr int NT  = 256;   // threads per workgroup (8 wave32)
constexpr int YS  = 72;    // row stride for [MT][64] state buffers (padded)
constexpr int HSs = 272;   // row stride for hidden buffer [MT][256] (padded)
constexpr int W1TS = 72;   // W1^T: [256 rows][K=64] padded stride
constexpr int W2TS = 264;  // W2^T: [64 rows][K=256] padded stride

// Hardware tanh (V_TANH_F32 is a CDNA5 TRANS op) instead of the branchy
// OCML exp-based expansion.
#if defined(__has_builtin)
#if __has_builtin(__builtin_amdgcn_tanhf)
#define HAVE_HW_TANH_BUILTIN 1
#endif
#endif

__device__ __forceinline__ float fast_tanh(float x) {
#if defined(HAVE_HW_TANH_BUILTIN)
    return __builtin_amdgcn_tanhf(x);
#elif defined(__gfx1250__)
    float r;
    // v_nop covers the TRANS-result hazard (1 independent op / V_NOP before
    // the output may be consumed), since the compiler can't see through asm.
    asm("v_tanh_f32 %0, %1\n\tv_nop" : "=v"(r) : "v"(x));
    return r;
#else
    return tanhf(x);
#endif
}

// f(y) = tanh(Y @ W1 + b1) @ W2 + b2 for one [MT x 64] tile, via f32 WMMA.
// yin/kout/hbuf/weights are all LDS pointers. Entry barrier covers yin,
// exit barrier publishes kout.
__device__ __forceinline__ void f_eval(const float* yin, float* kout,
                                       const float* w1t, const float* w2t,
                                       const float* b1, const float* b2,
                                       float* hbuf, int tid)
{
    __syncthreads();                       // yin fully built
    const int lane = tid & 31, wave = tid >> 5;
    const int half = lane >> 4, l16 = lane & 15;
    const int mt = wave >> 2, wq = wave & 3;   // row-tile (0..1), col-group (0..3)
    const int am = mt * 16 + l16;              // A-fragment row for this lane

    // ---- GEMM1: H = tanh(Yin[32x64] @ W1[64x256] + b1), 4 col-tiles per wave
    v8f acc[4];
    int ncol[4];
#pragma unroll
    for (int j = 0; j < 4; ++j) {
        ncol[j] = (wq * 4 + j) * 16 + l16;
        const float bv = b1[ncol[j]];
#pragma unroll
        for (int r = 0; r < 8; ++r) acc[j][r] = bv;
    }
#pragma unroll 4
    for (int kb = 0; kb < 16; ++kb) {      // K = 64 in steps of 4
        const v2f a = *(const v2f*)&yin[am * YS + kb * 4 + 2 * half];
#pragma unroll
        for (int j = 0; j < 4; ++j) {
            const v2f b = *(const v2f*)&w1t[ncol[j] * W1TS + kb * 4 + 2 * half];
            acc[j] = __builtin_amdgcn_wmma_f32_16x16x4_f32(
                false, a, false, b, (short)0, acc[j], false, false);
        }
    }
#pragma unroll
    for (int j = 0; j < 4; ++j)
#pragma unroll
        for (int r = 0; r < 8; ++r) {
            const int mm = mt * 16 + 8 * half + r;
            hbuf[mm * HSs + ncol[j]] = fast_tanh(acc[j][r]);
        }
    __syncthreads();                       // H fully built

    // ---- GEMM2: K = H[32x256] @ W2[256x64] + b2, 1 tile per wave
    const int n2 = wq * 16 + l16;
    v8f c;
    {
        const float bv = b2[n2];
#pragma unroll
        for (int r = 0; r < 8; ++r) c[r] = bv;
    }
#pragma unroll 4
    for (int kb = 0; kb < 64; ++kb) {      // K = 256 in steps of 4
        const v2f a = *(const v2f*)&hbuf[am * HSs + kb * 4 + 2 * half];
        const v2f b = *(const v2f*)&w2t[n2 * W2TS + kb * 4 + 2 * half];
        c = __builtin_amdgcn_wmma_f32_16x16x4_f32(
            false, a, false, b, (short)0, c, false, false);
    }
#pragma unroll
    for (int r = 0; r < 8; ++r)
        kout[(mt * 16 + 8 * half + r) * YS + n2] = c[r];
    __syncthreads();                       // kout published
}

__global__ __launch_bounds__(NT, 1)
void node_dopri5_kernel(const float* __restrict__ y0, const float* __restrict__ t,
                        const float* __restrict__ W1, const float* __restrict__ b1g,
                        const float* __restrict__ W2, const float* __restrict__ b2g,
                        float* __restrict__ out)
{
    __shared__ __align__(16) float sW1t[256 * W1TS];   // W1 transposed [n][k]
    __shared__ __align__(16) float sW2t[64 * W2TS];    // W2 transposed [n][k]
    __shared__ __align__(16) float sB1[256];
    __shared__ __align__(16) float sB2[64];
    __shared__ __align__(16) float sH[MT * HSs];
    __shared__ __align__(16) float sY[MT * YS];
    __shared__ __align__(16) float sYi[MT * YS];
    __shared__ __align__(16) float sK[6][MT * YS];

    const int tid = threadIdx.x;
    const int mb  = blockIdx.x * MT;       // batch-row base for this WG

    // Stage weights into LDS (transposed so B-fragments are single b64 loads)
    for (int i = tid; i < 64 * 256; i += NT) {
        const int k = i >> 8, n = i & 255;            // W1[k][n]
        sW1t[n * W1TS + k] = W1[i];
    }
    for (int i = tid; i < 256 * 64; i += NT) {
        const int k = i >> 6, n = i & 63;             // W2[k][n]
        sW2t[n * W2TS + k] = W2[i];
    }
    for (int i = tid; i < 256; i += NT) sB1[i] = b1g[i];
    for (int i = tid; i < 64;  i += NT) sB2[i] = b2g[i];

    // Load initial state tile; trajectory includes y0 at t[0]
    for (int e = tid; e < MT * 64; e += NT) {
        const int m = e >> 6, n = e & 63;
        const float v = y0[(size_t)(mb + m) * D_ + n];
        sY[m * YS + n] = v;
        out[(size_t)(mb + m) * D_ + n] = v;
    }
    __syncthreads();

    // Dormand-Prince 5(4) coefficients (5th-order row)
    const float A21 = 1.f/5.f;
    const float A31 = 3.f/40.f,        A32 = 9.f/40.f;
    const float A41 = 44.f/45.f,       A42 = -56.f/15.f,     A43 = 32.f/9.f;
    const float A51 = 19372.f/6561.f,  A52 = -25360.f/2187.f, A53 = 64448.f/6561.f, A54 = -212.f/729.f;
    const float A61 = 9017.f/3168.f,   A62 = -355.f/33.f,     A63 = 46732.f/5247.f, A64 = 49.f/176.f, A65 = -5103.f/18656.f;
    const float B1c = 35.f/384.f, B3c = 500.f/1113.f, B4c = 125.f/192.f,
                B5c = -2187.f/6784.f, B6c = 11.f/84.f;

    for (int it = 0; it < T_ - 1; ++it) {
        const float dt = (t[it + 1] - t[it]) * (1.0f / SUB);
        for (int s = 0; s < SUB; ++s) {
            // stage 0: yi = y
            for (int e = tid; e < MT * 64; e += NT) {
                const int o = (e >> 6) * YS + (e & 63);
                sYi[o] = sY[o];
            }
            f_eval(sYi, sK[0], sW1t, sW2t, sB1, sB2, sH, tid);
            // stage 1
            for (int e = tid; e < MT * 64; e += NT) {
                const int o = (e >> 6) * YS + (e & 63);
                sYi[o] = sY[o] + dt * (A21 * sK[0][o]);
            }
            f_eval(sYi, sK[1], sW1t, sW2t, sB1, sB2, sH, tid);
            // stage 2
            for (int e = tid; e < MT * 64; e += NT) {
                const int o = (e >> 6) * YS + (e & 63);
                sYi[o] = sY[o] + dt * (A31 * sK[0][o] + A32 * sK[1][o]);
            }
            f_eval(sYi, sK[2], sW1t, sW2t, sB1, sB2, sH, tid);
            // stage 3
            for (int e = tid; e < MT * 64; e += NT) {
                const int o = (e >> 6) * YS + (e & 63);
                sYi[o] = sY[o] + dt * (A41 * sK[0][o] + A42 * sK[1][o] + A43 * sK[2][o]);
            }
            f_eval(sYi, sK[3], sW1t, sW2t, sB1, sB2, sH, tid);
            // stage 4
            for (int e = tid; e < MT * 64; e += NT) {
                const int o = (e >> 6) * YS + (e & 63);
                sYi[o] = sY[o] + dt * (A51 * sK[0][o] + A52 * sK[1][o]
                                      + A53 * sK[2][o] + A54 * sK[3][o]);
            }
            f_eval(sYi, sK[4], sW1t, sW2t, sB1, sB2, sH, tid);
            // stage 5
            for (int e = tid; e < MT * 64; e += NT) {
                const int o = (e >> 6) * YS + (e & 63);
                sYi[o] = sY[o] + dt * (A61 * sK[0][o] + A62 * sK[1][o]
                                      + A63 * sK[2][o] + A64 * sK[3][o]
                                      + A65 * sK[4][o]);
            }
            f_eval(sYi, sK[5], sW1t, sW2t, sB1, sB2, sH, tid);
            // combine (B-row; B2 == 0)
            for (int e = tid; e < MT * 64; e += NT) {
                const int o = (e >> 6) * YS + (e & 63);
                sY[o] += dt * (B1c * sK[0][o] + B3c * sK[2][o] + B4c * sK[3][o]
                              + B5c * sK[4][o] + B6c * sK[5][o]);
            }
            // same thread wrote sY[o] as will read it in the next stage-0 copy;
            // f_eval's entry barrier covers cross-thread visibility.
        }
        // emit trajectory sample at t[it+1]
        for (int e = tid; e < MT * 64; e += NT) {
            const int m = e >> 6, n = e & 63;
            out[(size_t)(it + 1) * B_ * D_ + (size_t)(mb + m) * D_ + n] = sY[m * YS + n];
        }
    }
}

extern "C" void kernel_launch(void* const* d_in, const int* in_sizes, int n_in,
                              void* d_out, int out_size, void* d_ws, size_t ws_size,
                              hipStream_t stream) {
    (void)in_sizes; (void)n_in; (void)out_size; (void)d_ws; (void)ws_size;
    const float* y0 = (const float*)d_in[0];
    const float* t  = (const float*)d_in[1];
    const float* W1 = (const float*)d_in[2];
    const float* b1 = (const float*)d_in[3];
    const float* W2 = (const float*)d_in[4];
    const float* b2 = (const float*)d_in[5];
    float* out = (float*)d_out;
    hipLaunchKernelGGL(node_dopri5_kernel, dim3(B_ / MT), dim3(NT), 0, stream,
                       y0, t, W1, b1, W2, b2, out);
}